// ConvolutionLayer_88424786690167
// MI455X (gfx1250) — compile-verified
//
#include <hip/hip_runtime.h>

typedef float v2f __attribute__((ext_vector_type(2)));
typedef float v8f __attribute__((ext_vector_type(8)));

#define D_FEAT 128
#define LDW (D_FEAT + 4)   // padded LDS row stride (132 floats): 528B rows stay
                           // 16B-aligned, breaks the stride-128 bank conflict

// ---------------------------------------------------------------------------
// Kernel 1: zero the aggregation workspace (must re-zero every call).
// ---------------------------------------------------------------------------
__global__ void zero_agg_kernel(float4* __restrict__ agg4, int n4) {
    int i = blockIdx.x * blockDim.x + threadIdx.x;
    if (i < n4) agg4[i] = make_float4(0.f, 0.f, 0.f, 0.f);
}

// ---------------------------------------------------------------------------
// Kernel 2: edge gather + scatter-add.  One wave32 per edge: lane l loads
// X[src][4l..4l+3] as float4 (coalesced 512B/wave, global_load_b128) and
// atomically adds into agg[dst][4l..4l+3] (global_atomic_add_f32, no return).
// X/agg are L2-resident (51.2 MB each vs 192 MB L2).
// ---------------------------------------------------------------------------
__global__ __launch_bounds__(256) void scatter_add_kernel(
    const float* __restrict__ X,
    const int*   __restrict__ eSrc,
    const int*   __restrict__ eDst,
    float*       __restrict__ agg,
    int nEdges) {
    int e = blockIdx.x * 8 + (threadIdx.x >> 5);   // 8 waves per block
    if (e >= nEdges) return;
    int lane = threadIdx.x & 31;
    int src = eSrc[e];
    int dst = eDst[e];
    const float4 v = ((const float4*)(X + (size_t)src * D_FEAT))[lane];
    float* d = agg + (size_t)dst * D_FEAT + lane * 4;
    atomicAdd(d + 0, v.x);
    atomicAdd(d + 1, v.y);
    atomicAdd(d + 2, v.z);
    atomicAdd(d + 3, v.w);
}

// ---------------------------------------------------------------------------
// Kernel 3: out = agg @ w + b via V_WMMA_F32_16X16X4_F32.
// Block = 128 threads = 4 waves; w (128x128) + bias staged in LDS once.
// Each wave computes TWO 16-row M-tiles (32 rows x 128 cols): the two tiles
// share every B-fragment DS load (halves LDS traffic per WMMA) and form two
// independent accumulator chains that interleave to hide XDL latency.
//
// fp32 WMMA operand layouts (ISA 7.12.2):
//   A 16x4:  lane m=lane&15, hi=lane>>4: comp0=A[m][2hi], comp1=A[m][2hi+1]
//   B 4x16:  lane n=lane&15, hi:         comp0=B[2hi][n], comp1=B[2hi+1][n]
//   C/D:     VGPR v holds D[v + 8*hi][n]
// ---------------------------------------------------------------------------
__global__ __launch_bounds__(128) void gemm_bias_kernel(
    const float* __restrict__ agg,
    const float* __restrict__ w,
    const float* __restrict__ bias,
    float*       __restrict__ out,
    int nRows) {
    __shared__ float lw[D_FEAT * LDW];
    __shared__ float lb[D_FEAT];

    int tid = threadIdx.x;

    // Stage w into padded LDS with float4 copies (row base 528B = 16B aligned).
    const float4* w4 = (const float4*)w;
    for (int i4 = tid; i4 < (D_FEAT * D_FEAT) / 4; i4 += 128) {
        int row  = i4 >> 5;        // 32 float4 per row
        int col4 = i4 & 31;
        ((float4*)(lw + row * LDW))[col4] = w4[i4];
    }
    if (tid < D_FEAT) lb[tid] = bias[tid];
    __syncthreads();

    int wave = tid >> 5;
    int lane = tid & 31;
    int m    = lane & 15;
    int hi   = lane >> 4;

    int tileM0 = (blockIdx.x * 4 + wave) * 32;   // first 16-row tile
    int tileM1 = tileM0 + 16;                    // second 16-row tile

    int rowA0 = tileM0 + m; if (rowA0 >= nRows) rowA0 = nRows - 1;
    int rowA1 = tileM1 + m; if (rowA1 >= nRows) rowA1 = nRows - 1;

    // Preload both A strips: 32 K-chunks of float2 per lane per tile.
    v2f af0[32], af1[32];
    const float* arow0 = agg + (size_t)rowA0 * D_FEAT + 2 * hi;
    const float* arow1 = agg + (size_t)rowA1 * D_FEAT + 2 * hi;
#pragma unroll
    for (int kk = 0; kk < 32; ++kk) {
        af0[kk] = *(const v2f*)(arow0 + 4 * kk);
        af1[kk] = *(const v2f*)(arow1 + 4 * kk);
    }

#pragma unroll 1
    for (int j = 0; j < 8; ++j) {
        int col = 16 * j + m;      // output / B column for this lane
        v8f c0 = {0.f, 0.f, 0.f, 0.f, 0.f, 0.f, 0.f, 0.f};
        v8f c1 = {0.f, 0.f, 0.f, 0.f, 0.f, 0.f, 0.f, 0.f};
#pragma unroll
        for (int kk = 0; kk < 32; ++kk) {
            v2f bf;
            bf.x = lw[(4 * kk + 2 * hi)     * LDW + col];
            bf.y = lw[(4 * kk + 2 * hi + 1) * LDW + col];
            // 8 args: (neg_a, A, neg_b, B, c_mod, C, reuse_a, reuse_b)
            c0 = __builtin_amdgcn_wmma_f32_16x16x4_f32(
                     false, af0[kk], false, bf, (short)0, c0, false, false);
            c1 = __builtin_amdgcn_wmma_f32_16x16x4_f32(
                     false, af1[kk], false, bf, (short)0, c1, false, false);
        }
        float bb = lb[col];
#pragma unroll
        for (int v = 0; v < 8; ++v) {
            int r0 = tileM0 + v + 8 * hi;
            int r1 = tileM1 + v + 8 * hi;
            if (r0 < nRows) out[(size_t)r0 * D_FEAT + col] = c0[v] + bb;
            if (r1 < nRows) out[(size_t)r1 * D_FEAT + col] = c1[v] + bb;
        }
    }
}

// ---------------------------------------------------------------------------
extern "C" void kernel_launch(void* const* d_in, const int* in_sizes, int n_in,
                              void* d_out, int out_size, void* d_ws, size_t ws_size,
                              hipStream_t stream) {
    const float* X    = (const float*)d_in[0];
    const int*   eSrc = (const int*)d_in[1];
    const int*   eDst = (const int*)d_in[2];
    const float* w    = (const float*)d_in[3];
    const float* b    = (const float*)d_in[4];
    float* out = (float*)d_out;
    float* agg = (float*)d_ws;          // nNodes * 128 floats (51.2 MB)

    int nNodes = in_sizes[0] / D_FEAT;
    int nEdges = in_sizes[1];

    int n4 = (nNodes * D_FEAT) / 4;
    zero_agg_kernel<<<(n4 + 255) / 256, 256, 0, stream>>>((float4*)agg, n4);

    scatter_add_kernel<<<(nEdges + 7) / 8, 256, 0, stream>>>(
        X, eSrc, eDst, agg, nEdges);

    int nBlocks = (nNodes + 127) / 128;   // 128 rows per block (4 waves x 32)
    gemm_bias_kernel<<<nBlocks, 128, 0, stream>>>(agg, w, b, out, nNodes);
}